// SigLoss_3272765080271
// MI455X (gfx1250) — compile-verified
//
#include <hip/hip_runtime.h>

// ---------------------------------------------------------------------------
// Signature-kernel MMD loss, fused batched-GEMM (WMMA f32 16x16x4) + Goursat
// PDE anti-diagonal wavefront, one workgroup per (pair-type, batch) problem.
// A=128 paths, M=512 points, D=16 dims, K-grid 512x512, inc 511x511.
// Wavefront carries K_{d-1}[t] and K_{d-2}[t-1] in registers: only 2 LDS
// loads + 1 LDS store + 1 barrier per diagonal on the critical path.
// ---------------------------------------------------------------------------

typedef __attribute__((ext_vector_type(2))) float v2f;
typedef __attribute__((ext_vector_type(8))) float v8f;

#define A_BATCH   128
#define MPTS      512            // points per path
#define DDIM      16             // path dimension
#define GRID_P    512            // K grid edge (Mi+1)
#define NROWS     511            // number of increments
#define RSTR      17             // padded LDS row stride for dX/dY (floats)
#define TSTR      17             // padded tile row stride (floats)
#define TILE_F    (16 * TSTR)    // floats per 16x16 tile (272)
#define NTILE     32             // tiles per anti-diagonal (512/16)

// shared-memory layout (float offsets)
#define OFF_DX    0
#define OFF_DY    (OFF_DX + GRID_P * RSTR)                 // 8704
#define OFF_TILE  (OFF_DY + GRID_P * RSTR)                 // 17408
#define OFF_K     (OFF_TILE + 2 * NTILE * TILE_F)          // +17408 = 34816
#define SMEM_F    (OFF_K + 2 * GRID_P)                     // 35840 floats
#define SMEM_B    (SMEM_F * 4)                             // 143360 bytes

__global__ __launch_bounds__(512, 1)
void sig_pde_kernel(const float* __restrict__ X, const float* __restrict__ Y,
                    float* __restrict__ ws) {
    extern __shared__ float sm[];
    float* dXl   = sm + OFF_DX;
    float* dYl   = sm + OFF_DY;
    float* tiles = sm + OFF_TILE;
    float* kbuf  = sm + OFF_K;

    const int p    = blockIdx.x;       // 0..383
    const int type = p >> 7;           // 0: xx, 1: yy, 2: xy
    const int a    = p & (A_BATCH - 1);
    const float coeff = (type == 2) ? -2.0f : 1.0f;

    const float* Abase = (type == 1 ? Y : X) + (size_t)a * MPTS * DDIM;
    const float* Bbase = (type == 0 ? X : Y) + (size_t)a * MPTS * DDIM;

    const int t = threadIdx.x;         // 0..511  (thread t owns grid row i=t)

    // ---- stage increment matrices dX (A side) and dY (B side) into LDS ----
    if (t < NROWS) {
        const float4* r0 = (const float4*)(Abase + (size_t)t * DDIM);
        const float4* r1 = (const float4*)(Abase + (size_t)(t + 1) * DDIM);
#pragma unroll
        for (int q = 0; q < 4; ++q) {
            float4 c0 = r0[q], c1 = r1[q];
            dXl[t * RSTR + 4 * q + 0] = c1.x - c0.x;
            dXl[t * RSTR + 4 * q + 1] = c1.y - c0.y;
            dXl[t * RSTR + 4 * q + 2] = c1.z - c0.z;
            dXl[t * RSTR + 4 * q + 3] = c1.w - c0.w;
        }
        r0 = (const float4*)(Bbase + (size_t)t * DDIM);
        r1 = (const float4*)(Bbase + (size_t)(t + 1) * DDIM);
#pragma unroll
        for (int q = 0; q < 4; ++q) {
            float4 c0 = r0[q], c1 = r1[q];
            dYl[t * RSTR + 4 * q + 0] = c1.x - c0.x;
            dYl[t * RSTR + 4 * q + 1] = c1.y - c0.y;
            dYl[t * RSTR + 4 * q + 2] = c1.z - c0.z;
            dYl[t * RSTR + 4 * q + 3] = c1.w - c0.w;
        }
    } else {                            // t == 511: zero pad row
#pragma unroll
        for (int q = 0; q < DDIM; ++q) {
            dXl[t * RSTR + q] = 0.0f;
            dYl[t * RSTR + q] = 0.0f;
        }
    }

    // ---- init wavefront state for diagonals d=0, d=1 ----
    float* bufPrev = kbuf;             // holds diag d-1 (starts: diag 1)
    float* bufCur  = kbuf + GRID_P;    // target for diag d
    float myK    = (t <= 1) ? 1.0f : 0.0f;   // K_{1}[t]   (register carry)
    float leftK2 = (t == 1) ? 1.0f : 0.0f;   // K_{0}[t-1] (register carry)
    bufPrev[t] = myK;
    __syncthreads();

    const int lane = t & 31;
    const int wv   = t >> 5;            // wave id 0..15
    const int m15  = lane & 15;
    const int kb2  = (lane >> 4) << 1;  // K pair base: 0 (lanes 0-15) / 2 (16-31)

    // one wavefront step for K-grid anti-diagonal d (reads inc diag s=d-2)
    auto diag_step = [&](int d) {
        const float leftK1 = (t > 0) ? bufPrev[t - 1] : 0.0f;  // K_{d-1}[t-1]
        const int j = d - t;
        float nv = 0.0f;
        if (j >= 0 && j < GRID_P) {
            if (t == 0 || j == 0) {
                nv = 1.0f;
            } else {
                const int ii = t - 1, jj = j - 1;
                const int ti = ii >> 4, tj = jj >> 4;
                const float incv =
                    tiles[(((ti + tj) & 1) * NTILE + ti) * TILE_F +
                          (ii & 15) * TSTR + (jj & 15)];
                nv = myK + leftK1 + leftK2 * (incv - 1.0f);
            }
        }
        bufCur[t] = nv;
        __syncthreads();
        leftK2 = leftK1;               // K_{d-1}[t-1] becomes next step's K_{d'-2}[t-1]
        myK    = nv;                   // this thread's K_{d}[t]
        float* tmp = bufPrev; bufPrev = bufCur; bufCur = tmp;
    };

    // ---- main loop over tile anti-diagonals c = ti+tj in [0,62] ----
    for (int c = 0; c < 63; ++c) {
        // (1) compute inc tiles of anti-diagonal c into ring bank c&1 via WMMA
        const int lo = (c > 31) ? (c - 31) : 0;
        const int hi = (c < 31) ? c : 31;
        for (int q = lo + wv; q <= hi; q += 16) {   // wave-uniform bounds
            const int ti = q, tj = c - q;
            const float* ar = dXl + (16 * ti + m15) * RSTR + kb2;
            const float* br = dYl + (16 * tj + m15) * RSTR + kb2;
            v8f acc = {0.0f, 0.0f, 0.0f, 0.0f, 0.0f, 0.0f, 0.0f, 0.0f};
#pragma unroll
            for (int kk = 0; kk < 4; ++kk) {        // accumulate K=16 as 4x K=4
                v2f av, bv;
                av.x = ar[4 * kk]; av.y = ar[4 * kk + 1];
                bv.x = br[4 * kk]; bv.y = br[4 * kk + 1];
                acc = __builtin_amdgcn_wmma_f32_16x16x4_f32(
                    false, av, false, bv, (short)0, acc, false, false);
            }
            // D layout: VGPR r -> (M=r, N=lane) lanes 0-15 ; (M=8+r) lanes 16-31
            float* tb = tiles + ((c & 1) * NTILE + ti) * TILE_F +
                        ((lane >= 16) ? 8 * TSTR : 0) + m15;
#pragma unroll
            for (int r = 0; r < 8; ++r) tb[r * TSTR] = acc[r];
        }
        __syncthreads();

        // (2) sweep 16 K-grid diagonals needing inc diags s in [16c, 16c+15];
        //     these touch only tile anti-diagonals c and c-1 (both resident)
        const int d0 = 16 * c + 2;
        for (int dd = 0; dd < 16; ++dd) diag_step(d0 + dd);
    }
    // tail: d = 1010..1022 (inc diags 1008..1020 live entirely in bank 62&1)
    for (int d = 1010; d <= 1022; ++d) diag_step(d);

    // after processing d=1022, K[511,511] is this thread's register carry
    if (t == GRID_P - 1) ws[p] = coeff * myK;
}

// fixed-order deterministic reduction: mean over batch of (kxx + kyy - 2 kxy)
__global__ void sig_reduce_kernel(const float* __restrict__ ws,
                                  float* __restrict__ out) {
    __shared__ float sm[128];
    const int t = threadIdx.x;
    sm[t] = ws[t] + ws[128 + t] + ws[256 + t];
    __syncthreads();
    for (int s = 64; s > 0; s >>= 1) {
        if (t < s) sm[t] += sm[t + s];
        __syncthreads();
    }
    if (t == 0) out[0] = sm[0] * (1.0f / 128.0f);
}

extern "C" void kernel_launch(void* const* d_in, const int* in_sizes, int n_in,
                              void* d_out, int out_size, void* d_ws, size_t ws_size,
                              hipStream_t stream) {
    const float* X = (const float*)d_in[0];
    const float* Y = (const float*)d_in[1];
    float* ws  = (float*)d_ws;    // 384 floats of per-problem results
    float* out = (float*)d_out;   // 1 float

    sig_pde_kernel<<<dim3(3 * A_BATCH), dim3(512), SMEM_B, stream>>>(X, Y, ws);
    sig_reduce_kernel<<<dim3(1), dim3(128), 0, stream>>>(ws, out);
}